// GumbelVectorQuantizer_14774687498251
// MI455X (gfx1250) — compile-verified
//
#include <hip/hip_runtime.h>

namespace {
constexpr int K_DIM     = 1024;  // D
constexpr int N_DIM     = 640;   // G*V
constexpr int V_NUM     = 320;   // vars per group
constexpr int VAR_DIM   = 128;   // VQ / G
constexpr int M_TILE    = 64;    // rows per workgroup
constexpr int NWAVES    = 16;    // 512 threads
constexpr int KC        = 128;   // K chunk staged in LDS
constexpr int XS_STRIDE = 132;   // 128 + 4 floats pad -> conflict-free ds_load_b64
constexpr int NT        = 10;    // 16-wide N tiles per wave (160 cols)

typedef float v2f __attribute__((ext_vector_type(2)));
typedef float v8f __attribute__((ext_vector_type(8)));
typedef int   v4i __attribute__((ext_vector_type(4)));
}

// CDNA5 async global->LDS staging path (ASYNCcnt-tracked), with fallback.
#if defined(__gfx1250__) && __has_builtin(__builtin_amdgcn_global_load_async_to_lds_b128)
#define GVQ_ASYNC_STAGE 1
typedef __attribute__((address_space(1))) v4i* gvq_gptr_t;
typedef __attribute__((address_space(3))) v4i* gvq_lptr_t;
#endif

__device__ __forceinline__ void gvq_wait_async0() {
#if defined(__gfx1250__)
#if __has_builtin(__builtin_amdgcn_s_wait_asynccnt)
    __builtin_amdgcn_s_wait_asynccnt(0);
#else
    asm volatile("s_wait_asynccnt 0x0" ::: "memory");
#endif
#endif
}

__global__ __launch_bounds__(512)
void gvq_fused(const float* __restrict__ x,
               const float* __restrict__ Wm,
               const float* __restrict__ bias,
               const float* __restrict__ codebook,
               const float* __restrict__ gumbels,
               float* __restrict__ out)
{
    __shared__ float xs[M_TILE * XS_STRIDE];
    __shared__ float pv[M_TILE][2][2];
    __shared__ int   pi[M_TILE][2][2];
    __shared__ int   choice[M_TILE * 2];

    const int tid  = threadIdx.x;
    const int wave = tid >> 5;
    const int lane = tid & 31;
    const int lo   = lane & 15;
    const int hi   = lane >> 4;
    const int row0 = blockIdx.x * M_TILE;

    const int m16 = (wave & 3) * 16;          // wave's M tile (4 tiles across 64 rows)
    const int n0  = (wave >> 2) * (NT * 16);  // wave's N range start (0/160/320/480)

    v8f acc[NT];
    #pragma unroll
    for (int t = 0; t < NT; ++t) {
        #pragma unroll
        for (int e = 0; e < 8; ++e) acc[t][e] = 0.0f;
    }

    // Per-lane fragment bases (ISA 7.12.2, f32 16x16x4):
    // A: lane L holds x[m16 + (L&15), k + 2*(L>>4) + {0,1}]
    // B: lane L holds W[n + (L&15),  k + 2*(L>>4) + {0,1}]  (W is [N,K] row-major)
    const float* wbase = Wm + (size_t)(n0 + lo) * K_DIM + 2 * hi;
    const float* xsa   = xs + (m16 + lo) * XS_STRIDE + 2 * hi;
    const float* xrow  = x  + (size_t)row0 * K_DIM;

    for (int kc = 0; kc < K_DIM; kc += KC) {
        __syncthreads();
        // Stage the 64x128 x-chunk: 2048 float4, 4 per thread.
        #pragma unroll
        for (int i = 0; i < 4; ++i) {
            const int f = tid + 512 * i;
            const int r = f >> 5;          // row 0..63
            const int s = f & 31;          // float4 segment 0..31
            const float* gp = xrow + (size_t)r * K_DIM + kc + s * 4;
            float*       lp = &xs[r * XS_STRIDE + s * 4];
#if defined(GVQ_ASYNC_STAGE)
            __builtin_amdgcn_global_load_async_to_lds_b128(
                (gvq_gptr_t)(size_t)gp, (gvq_lptr_t)lp, 0, 0);
#else
            const float4 v4 = *(const float4*)gp;
            lp[0] = v4.x; lp[1] = v4.y; lp[2] = v4.z; lp[3] = v4.w;
#endif
        }
#if defined(GVQ_ASYNC_STAGE)
        gvq_wait_async0();
#endif
        __syncthreads();

        const float* wk = wbase + kc;
        if (kc + KC < K_DIM) {
            #pragma unroll
            for (int t = 0; t < NT; ++t)
                __builtin_prefetch(wk + KC + t * (16 * K_DIM), 0, 3);
        }

        #pragma unroll 4
        for (int kk = 0; kk < KC / 4; ++kk) {
            const int kb = kk * 4;
            const v2f a = *(const v2f*)(xsa + kb);          // ds_load (2addr-fused)
            #pragma unroll
            for (int t = 0; t < NT; ++t) {
                const v2f b = *(const v2f*)(wk + t * (16 * K_DIM) + kb); // global_load_b64 (L2)
                acc[t] = __builtin_amdgcn_wmma_f32_16x16x4_f32(
                    false, a, false, b, (short)0, acc[t], false, false);
            }
        }
    }

    // Epilogue: logits + bias + gumbel, per-row argmax over this wave's 160 cols.
    // C layout: VGPR j, lanes 0-15 -> (row m16+j, col n0+t*16+lane), lanes 16-31 -> row m16+j+8.
    const int g    = n0 / V_NUM;            // group 0/1
    const int half = (n0 / (NT * 16)) & 1;  // which 160-col half of the group
    #pragma unroll
    for (int j = 0; j < 8; ++j) {
        const int rl = m16 + j + 8 * hi;
        const size_t grow = (size_t)(row0 + rl) * N_DIM;
        float bv = -3.402823466e+38f;
        int   bi = 0;
        #pragma unroll
        for (int t = 0; t < NT; ++t) {
            const int col = n0 + t * 16 + lo;
            const float val = acc[t][j] + bias[col] + gumbels[grow + col];
            if (val > bv) { bv = val; bi = col - g * V_NUM; } // idx ascending in t -> first-max kept
        }
        // combine across the 16 lanes of this half (xor masks 1..8 stay within the half)
        #pragma unroll
        for (int off = 1; off <= 8; off <<= 1) {
            const float ov = __shfl_xor(bv, off, 32);
            const int   oi = __shfl_xor(bi, off, 32);
            if (ov > bv || (ov == bv && oi < bi)) { bv = ov; bi = oi; }
        }
        if (lo == j) { pv[rl][g][half] = bv; pi[rl][g][half] = bi; }
    }
    __syncthreads();

    // Merge the two 160-col halves per (row, group). half0 indices < half1 indices,
    // so strict '>' implements lowest-index tie-break (jnp.argmax semantics).
    if (tid < M_TILE * 2) {
        const int rl = tid >> 1, gg = tid & 1;
        const float v0 = pv[rl][gg][0], v1 = pv[rl][gg][1];
        choice[tid] = (v1 > v0) ? pi[rl][gg][1] : pi[rl][gg][0];
    }
    __syncthreads();

    // Gather: one wave per (row, group) pair; 32 lanes x float4 = one 128-float codebook row.
    for (int p = wave; p < M_TILE * 2; p += NWAVES) {
        const int rl = p >> 1, gg = p & 1;
        const int idx = choice[p];
        const float4 cv = *(const float4*)(codebook +
                             (size_t)(gg * V_NUM + idx) * VAR_DIM + lane * 4);
        *(float4*)(out + (size_t)(row0 + rl) * (2 * VAR_DIM) + gg * VAR_DIM + lane * 4) = cv;
    }
}

extern "C" void kernel_launch(void* const* d_in, const int* in_sizes, int n_in,
                              void* d_out, int out_size, void* d_ws, size_t ws_size,
                              hipStream_t stream) {
    (void)in_sizes; (void)n_in; (void)out_size; (void)d_ws; (void)ws_size;
    const float* x        = (const float*)d_in[0];
    const float* Wm       = (const float*)d_in[1];
    const float* bias     = (const float*)d_in[2];
    const float* codebook = (const float*)d_in[3];
    const float* gumbels  = (const float*)d_in[4];
    float* out = (float*)d_out;

    const int M_TOTAL = 32 * 2048;            // B*T = 65536 rows
    dim3 grid(M_TOTAL / M_TILE);              // 1024 workgroups
    dim3 block(512);                          // 16 wave32s
    gvq_fused<<<grid, block, 0, stream>>>(x, Wm, bias, codebook, gumbels, out);
}